// GCN_61091614819158
// MI455X (gfx1250) — compile-verified
//
#include <hip/hip_runtime.h>
#include <hip/hip_bf16.h>
#include <math.h>

typedef __attribute__((ext_vector_type(2))) float v2f;
typedef __attribute__((ext_vector_type(8))) float v8f;

#define HIDDIM 128

// ---------------- CSR build ----------------

__global__ __launch_bounds__(256) void k_zero2(int* a, int* b, int n) {
    int i = blockIdx.x * 256 + threadIdx.x;
    if (i < n) { a[i] = 0; b[i] = 0; }
}

__global__ __launch_bounds__(256) void k_count(const int* __restrict__ dst, int* __restrict__ cnt, int E) {
    int e = blockIdx.x * 256 + threadIdx.x;
    if (e < E) atomicAdd(&cnt[dst[e]], 1);
}

// per-block exclusive scan of cnt into rowptr; block sums out
__global__ __launch_bounds__(256) void k_scan1(const int* __restrict__ cnt, int* __restrict__ rowptr,
                                               int* __restrict__ bsum, int n) {
    __shared__ int sm[256];
    int i = blockIdx.x * 256 + threadIdx.x;
    int v = (i < n) ? cnt[i] : 0;
    sm[threadIdx.x] = v;
    __syncthreads();
    for (int off = 1; off < 256; off <<= 1) {
        int t = (threadIdx.x >= off) ? sm[threadIdx.x - off] : 0;
        __syncthreads();
        sm[threadIdx.x] += t;
        __syncthreads();
    }
    if (i < n) rowptr[i] = sm[threadIdx.x] - v;   // exclusive within block
    if (threadIdx.x == 255) bsum[blockIdx.x] = sm[255];
}

// single-block exclusive scan of block sums (nb <= 512)
__global__ __launch_bounds__(512) void k_scan2(int* bsum, int nb) {
    __shared__ int sm[512];
    int t = threadIdx.x;
    int v = (t < nb) ? bsum[t] : 0;
    sm[t] = v;
    __syncthreads();
    for (int off = 1; off < 512; off <<= 1) {
        int u = (t >= off) ? sm[t - off] : 0;
        __syncthreads();
        sm[t] += u;
        __syncthreads();
    }
    if (t < nb) bsum[t] = sm[t] - v;
}

// add block offsets; also deg = cnt+1 (self loop) -> dinv
__global__ __launch_bounds__(256) void k_scan3(int* __restrict__ rowptr, const int* __restrict__ bsum,
                                               const int* __restrict__ cnt, float* __restrict__ dinv,
                                               int n, int E) {
    int i = blockIdx.x * 256 + threadIdx.x;
    if (i < n) {
        rowptr[i] += bsum[blockIdx.x];
        dinv[i] = rsqrtf((float)(cnt[i] + 1));
    }
    if (i == 0) rowptr[n] = E;
}

__global__ __launch_bounds__(256) void k_fill(const int* __restrict__ src, const int* __restrict__ dst,
                                              const int* __restrict__ rowptr, int* __restrict__ fill,
                                              int* __restrict__ colidx, int E) {
    int e = blockIdx.x * 256 + threadIdx.x;
    if (e < E) {
        int d = dst[e];
        int p = atomicAdd(&fill[d], 1);
        colidx[rowptr[d] + p] = src[e];
    }
}

// ---------------- WMMA GEMM: out[N,128] = A[N,128] @ W[128,128]^T, epilogue scale/bias ----------------
// Each block: 128 rows, 8 waves; wave w computes rows [base+16w, base+16w+16) x 128 cols.
// f32 WMMA 16x16x4:
//   A frag v2f : lanes 0-15 -> (M=lane, K=kk+0/kk+1), lanes 16-31 -> (M=lane-16, K=kk+2/kk+3)
//   B frag v2f : mirrored K split, N = lane%16
//   C v8f      : VGPR p -> row p (lanes 0-15) / row p+8 (lanes 16-31), col = lane%16
// LDS holds W pre-swizzled into K-pair fragments so each B operand is ONE aligned ds_load_b64:
//   sWp[k2*256 + 2*j + h] = W[j][2*k2 + h]

__global__ __launch_bounds__(256) void k_gemm128(const float* __restrict__ A,
                                                 const float* __restrict__ W,
                                                 const float* __restrict__ bias,      // may be null
                                                 const float* __restrict__ rowscale,  // may be null
                                                 float* __restrict__ out, int N) {
    __shared__ float sWp[HIDDIM * HIDDIM];  // 64 KB, K-pair swizzled
    for (int idx = threadIdx.x; idx < HIDDIM * HIDDIM; idx += 256) {
        int k2 = idx >> 8;       // K pair index (0..63)
        int r  = idx & 255;
        int j  = r >> 1;         // output column (W row)
        int h  = r & 1;          // element within pair
        sWp[idx] = W[j * HIDDIM + 2 * k2 + h];
    }
    __syncthreads();

    int lane = threadIdx.x & 31;
    int wave = threadIdx.x >> 5;
    int rbase = blockIdx.x * 128 + wave * 16;
    int lo = lane & 15;
    int hi = lane >> 4;          // 0 for lanes 0-15, 1 for lanes 16-31
    int kh = hi << 1;            // K offset 0 or 2
    int r = rbase + lo;
    int rc = (r < N) ? r : (N - 1);
    const float* Arow = A + (size_t)rc * HIDDIM;

    // per-lane B base: half-wave picks k2 parity, column = jt*16+lo
    const float* Bbase = &sWp[hi * 256 + lo * 2];

    v8f acc[8];
#pragma unroll
    for (int jt = 0; jt < 8; jt++) acc[jt] = (v8f){0.f, 0.f, 0.f, 0.f, 0.f, 0.f, 0.f, 0.f};

    for (int kk = 0; kk < HIDDIM; kk += 4) {
        v2f a = *(const v2f*)(Arow + kk + kh);
        const float* bp = Bbase + (kk >> 1) * 256;   // (kk/2 + hi) pair block, hi folded into Bbase
#pragma unroll
        for (int jt = 0; jt < 8; jt++) {
            v2f b = *(const v2f*)(bp + jt * 32);
            acc[jt] = __builtin_amdgcn_wmma_f32_16x16x4_f32(
                false, a, false, b, (short)0, acc[jt], false, false);
        }
    }

    int mofs = hi * 8;
#pragma unroll
    for (int p = 0; p < 8; p++) {
        int row = rbase + mofs + p;
        if (row < N) {
            float rs = rowscale ? rowscale[row] : 1.0f;
#pragma unroll
            for (int jt = 0; jt < 8; jt++) {
                float v = acc[jt][p] * rs;
                if (bias) v += bias[jt * 16 + lo];
                out[(size_t)row * HIDDIM + jt * 16 + lo] = v;
            }
        }
    }
}

// ---------------- Aggregation: out = relu(dinv[i] * (hp[i] + sum_{j in N(i)} hp[j]) + b) ----------------
// one wave per node; lane owns 4 contiguous floats (32 lanes * 4 = 128); x4 unrolled gather

__global__ __launch_bounds__(256) void k_agg(const float* __restrict__ hp,
                                             const int* __restrict__ rowptr,
                                             const int* __restrict__ colidx,
                                             const float* __restrict__ dinv,
                                             const float* __restrict__ bias,
                                             float* __restrict__ out, int N) {
    int wave = threadIdx.x >> 5;
    int lane = threadIdx.x & 31;
    int node = blockIdx.x * 8 + wave;
    if (node >= N) return;
    const float4* hp4 = (const float4*)hp;
    float4 a = hp4[(size_t)node * 32 + lane];  // self-loop contribution
    int beg = rowptr[node];
    int end = rowptr[node + 1];
    int e = beg;
    for (; e + 3 < end; e += 4) {
        int s0 = colidx[e + 0];
        int s1 = colidx[e + 1];
        int s2 = colidx[e + 2];
        int s3 = colidx[e + 3];
        float4 m0 = hp4[(size_t)s0 * 32 + lane];
        float4 m1 = hp4[(size_t)s1 * 32 + lane];
        float4 m2 = hp4[(size_t)s2 * 32 + lane];
        float4 m3 = hp4[(size_t)s3 * 32 + lane];
        a.x += m0.x + m1.x + m2.x + m3.x;
        a.y += m0.y + m1.y + m2.y + m3.y;
        a.z += m0.z + m1.z + m2.z + m3.z;
        a.w += m0.w + m1.w + m2.w + m3.w;
    }
    for (; e < end; e++) {
        int s = colidx[e];
        float4 m = hp4[(size_t)s * 32 + lane];
        a.x += m.x; a.y += m.y; a.z += m.z; a.w += m.w;
    }
    float di = dinv[node];
    float4 b = ((const float4*)bias)[lane];
    float4 o;
    o.x = fmaxf(fmaf(a.x, di, b.x), 0.f);
    o.y = fmaxf(fmaf(a.y, di, b.y), 0.f);
    o.z = fmaxf(fmaf(a.z, di, b.z), 0.f);
    o.w = fmaxf(fmaf(a.w, di, b.w), 0.f);
    ((float4*)out)[(size_t)node * 32 + lane] = o;
}

// ---------------- Decoder + log_softmax: one thread per row ----------------

__global__ __launch_bounds__(256) void k_decoder(const float* __restrict__ x,
                                                 const float* __restrict__ Wd,
                                                 const float* __restrict__ bd,
                                                 float* __restrict__ out, int N) {
    __shared__ float sW[40 * HIDDIM];
    __shared__ float sb[40];
    for (int idx = threadIdx.x; idx < 40 * HIDDIM; idx += 256) sW[idx] = Wd[idx];
    if (threadIdx.x < 40) sb[threadIdx.x] = bd[threadIdx.x];
    __syncthreads();

    int row = blockIdx.x * 256 + threadIdx.x;
    if (row >= N) return;
    const float* xr = x + (size_t)row * HIDDIM;

    float acc[40];
#pragma unroll
    for (int j = 0; j < 40; j++) acc[j] = 0.f;

    for (int kc = 0; kc < 4; kc++) {
        float xv[32];
        const float4* xp = (const float4*)(xr + kc * 32);
#pragma unroll
        for (int t = 0; t < 8; t++) {
            float4 q = xp[t];
            xv[4 * t + 0] = q.x; xv[4 * t + 1] = q.y;
            xv[4 * t + 2] = q.z; xv[4 * t + 3] = q.w;
        }
        for (int j = 0; j < 40; j++) {
            const float* wr = &sW[j * HIDDIM + kc * 32];
            float s = 0.f;
#pragma unroll
            for (int t = 0; t < 32; t++) s = fmaf(xv[t], wr[t], s);
            acc[j] += s;
        }
    }

    float m = -1e30f;
#pragma unroll
    for (int j = 0; j < 40; j++) { acc[j] += sb[j]; m = fmaxf(m, acc[j]); }
    float ssum = 0.f;
#pragma unroll
    for (int j = 0; j < 40; j++) ssum += expf(acc[j] - m);
    float ls = logf(ssum);
#pragma unroll
    for (int j = 0; j < 40; j++) out[(size_t)row * 40 + j] = acc[j] - m - ls;
}

// ---------------- host side ----------------

extern "C" void kernel_launch(void* const* d_in, const int* in_sizes, int n_in,
                              void* d_out, int out_size, void* d_ws, size_t ws_size,
                              hipStream_t stream) {
    const float* X    = (const float*)d_in[0];
    const int*   edge = (const int*)d_in[1];
    const float* encW = (const float*)d_in[2];
    const float* encb = (const float*)d_in[3];
    const float* W1   = (const float*)d_in[4];
    const float* b1   = (const float*)d_in[5];
    const float* W2   = (const float*)d_in[6];
    const float* b2   = (const float*)d_in[7];
    const float* decW = (const float*)d_in[8];
    const float* decb = (const float*)d_in[9];

    int N = in_sizes[0] / HIDDIM;
    int E = in_sizes[1] / 2;
    const int* src = edge;
    const int* dst = edge + E;

    // workspace layout
    float* xA     = (float*)d_ws;                 // N*128
    float* xB     = xA + (size_t)N * HIDDIM;      // N*128
    float* dinv   = xB + (size_t)N * HIDDIM;      // N
    int*   cnt    = (int*)(dinv + N);             // N
    int*   rowptr = cnt + N;                      // N+1
    int*   fill   = rowptr + (N + 1);             // N
    int*   colidx = fill + N;                     // E
    int*   bsum   = colidx + E;                   // <=512

    int nbN = (N + 255) / 256;
    int nbE = (E + 255) / 256;

    k_zero2<<<nbN, 256, 0, stream>>>(cnt, fill, N);
    k_count<<<nbE, 256, 0, stream>>>(dst, cnt, E);
    k_scan1<<<nbN, 256, 0, stream>>>(cnt, rowptr, bsum, N);
    k_scan2<<<1, 512, 0, stream>>>(bsum, nbN);
    k_scan3<<<nbN, 256, 0, stream>>>(rowptr, bsum, cnt, dinv, N, E);
    k_fill<<<nbE, 256, 0, stream>>>(src, dst, rowptr, fill, colidx, E);

    int gG = (N + 127) / 128;
    // encoder: xA = X @ encW^T + encb
    k_gemm128<<<gG, 256, 0, stream>>>(X, encW, encb, nullptr, xA, N);
    // conv1: h' = (xA @ W1^T) * dinv ; agg -> relu(dinv*agg + b1)
    k_gemm128<<<gG, 256, 0, stream>>>(xA, W1, nullptr, dinv, xB, N);
    k_agg<<<(N + 7) / 8, 256, 0, stream>>>(xB, rowptr, colidx, dinv, b1, xA, N);
    // conv2
    k_gemm128<<<gG, 256, 0, stream>>>(xA, W2, nullptr, dinv, xB, N);
    k_agg<<<(N + 7) / 8, 256, 0, stream>>>(xB, rowptr, colidx, dinv, b2, xA, N);
    // decoder + log_softmax
    k_decoder<<<nbN, 256, 0, stream>>>(xA, decW, decb, (float*)d_out, N);
}